// SparseEquivariantLayerBlock_73718818668668
// MI455X (gfx1250) — compile-verified
//
#include <hip/hip_runtime.h>

// Problem constants (from reference): DIN=DOUT=16, N=2048, NPARAMS=15
#define NSIDE   2048
#define NELEM   (NSIDE * NSIDE)       // 4194304 per channel
#define NCH     16                    // din == dout == 16
#define TTILES  (NSIDE / 16)          // 128 tiles per dimension
#define NPAIRS  (TTILES * (TTILES + 1) / 2)  // 8256 unordered tile pairs

typedef float v2f __attribute__((ext_vector_type(2)));
typedef float v8f __attribute__((ext_vector_type(8)));

// ---------------------------------------------------------------------------
// Stats pass 1: per-row sums + diagonal.  One wave per (channel,row); fixed
// accumulation order -> deterministic (no float atomics anywhere).
// grid = 16*256 blocks, 256 threads (8 waves -> 8 rows / block)
// ---------------------------------------------------------------------------
__global__ void rowsum_diag_kernel(const float* __restrict__ x,
                                   float* __restrict__ rowsum_t,   // [p][c]
                                   float* __restrict__ diag_t) {   // [p][c]
    const int bid  = blockIdx.x;
    const int c    = bid >> 8;
    const int rg   = bid & 255;
    const int lane = threadIdx.x & 31;
    const int wave = threadIdx.x >> 5;
    const int r    = rg * 8 + wave;
    const float* xr = x + (size_t)c * NELEM + (size_t)r * NSIDE;

    float acc = 0.f;
#pragma unroll
    for (int k = 0; k < 16; ++k) {
        float4 v = *reinterpret_cast<const float4*>(xr + lane * 4 + k * 128);
        acc += v.x + v.y + v.z + v.w;
    }
#pragma unroll
    for (int off = 16; off > 0; off >>= 1) acc += __shfl_down(acc, off, 32);
    if (lane == 0) {
        rowsum_t[r * NCH + c] = acc;
        diag_t[r * NCH + c]   = xr[r];   // x[c][r][r]
    }
}

// ---------------------------------------------------------------------------
// Stats pass 2: column-sum partials.  Thread owns one column over a 256-row
// chunk (fully coalesced 1KB lines), fixed order -> deterministic.
// grid = 16c * 8 schunk * 8 rchunk = 1024 blocks, 256 threads
// ---------------------------------------------------------------------------
__global__ void colsum_kernel(const float* __restrict__ x,
                              float* __restrict__ cpart) {  // [rc][s][c]
    const int bid = blockIdx.x;
    const int c   = bid >> 6;
    const int sc  = (bid >> 3) & 7;
    const int rc  = bid & 7;
    const int s   = sc * 256 + threadIdx.x;
    const float* xc = x + (size_t)c * NELEM;
    const int rbase = rc * 256;
    float acc = 0.f;
    for (int rr = 0; rr < 256; ++rr)
        acc += xc[(size_t)(rbase + rr) * NSIDE + s];
    cpart[((size_t)rc * NSIDE + s) * NCH + c] = acc;
}

// grid = 128 blocks * 256 threads = 32768 = 2048*16
__global__ void colsum_fin_kernel(const float* __restrict__ cpart,
                                  float* __restrict__ colsum_t) { // [p][c]
    const int g = blockIdx.x * blockDim.x + threadIdx.x;
    float acc = 0.f;
#pragma unroll
    for (int rc = 0; rc < 8; ++rc)
        acc += cpart[(size_t)rc * (NSIDE * NCH) + g];
    colsum_t[g] = acc;
}

// ---------------------------------------------------------------------------
// Scalars: trace[c], total[c] -> diag_scalars[o], const_terms[o]+bias.sum()
// single block, 256 threads, fixed-order reductions
// ---------------------------------------------------------------------------
__global__ void scal_kernel(const float* __restrict__ diag_t,
                            const float* __restrict__ rowsum_t,
                            const float* __restrict__ W,
                            const float* __restrict__ bias,
                            float* __restrict__ scal) {  // [0..15]=dscal, [16..31]=cconst+biassum
    __shared__ float red0[16][16], red1[16][16];
    __shared__ float str[16], sto[16];
    const int tid = threadIdx.x;
    const int c = tid & 15, seg = tid >> 4;
    float a = 0.f, b = 0.f;
    for (int p = seg; p < NSIDE; p += 16) {
        a += diag_t[p * NCH + c];
        b += rowsum_t[p * NCH + c];
    }
    red0[seg][c] = a; red1[seg][c] = b;
    __syncthreads();
    if (tid < 16) {
        float ta = 0.f, tb = 0.f;
#pragma unroll
        for (int s = 0; s < 16; ++s) { ta += red0[s][tid]; tb += red1[s][tid]; }
        str[tid] = ta; sto[tid] = tb;
    }
    __syncthreads();
    if (tid < 16) {
        const int o = tid;
        float ds = 0.f, cc = 0.f;
#pragma unroll
        for (int ci = 0; ci < 16; ++ci) {
            ds += W[5 * 256 + ci * 16 + o] * str[ci] + W[13 * 256 + ci * 16 + o] * sto[ci];
            cc += W[6 * 256 + ci * 16 + o] * str[ci] + W[14 * 256 + ci * 16 + o] * sto[ci];
        }
        float bs = 0.f;
#pragma unroll
        for (int k = 0; k < 15; ++k) bs += bias[k];
        scal[o]      = ds;        // diag_scalars[o]
        scal[16 + o] = cc + bs;   // const_terms[o] + sum(bias)
    }
}

// ---------------------------------------------------------------------------
// Build folded term arrays R[p][o], C[p][o], Dg[p][o]
// grid = 128 blocks * 256 threads = 32768 = 2048*16
// ---------------------------------------------------------------------------
__global__ void terms_kernel(const float* __restrict__ diag_t,
                             const float* __restrict__ rowsum_t,
                             const float* __restrict__ colsum_t,
                             const float* __restrict__ W,
                             const float* __restrict__ scal,
                             float* __restrict__ Rrow,
                             float* __restrict__ Ccol,
                             float* __restrict__ Dg) {
    const int g = blockIdx.x * blockDim.x + threadIdx.x;
    const int p = g >> 4, o = g & 15;
    float dterm = 0.f, rterm = 0.f, cterm = 0.f;
#pragma unroll
    for (int c = 0; c < 16; ++c) {
        const float dv = diag_t[p * NCH + c];
        const float rv = rowsum_t[p * NCH + c];
        const float cv = colsum_t[p * NCH + c];
        dterm += W[0 * 256 + c * 16 + o] * dv + W[3 * 256 + c * 16 + o] * rv + W[4 * 256 + c * 16 + o] * cv;
        rterm += W[1 * 256 + c * 16 + o] * dv + W[9 * 256 + c * 16 + o] * rv + W[11 * 256 + c * 16 + o] * cv;
        cterm += W[2 * 256 + c * 16 + o] * dv + W[10 * 256 + c * 16 + o] * rv + W[12 * 256 + c * 16 + o] * cv;
    }
    Rrow[p * NCH + o] = rterm + scal[16 + o];
    Ccol[p * NCH + o] = cterm;
    Dg[p * NCH + o]   = dterm + scal[o];
}

// ---------------------------------------------------------------------------
// Main kernel: per unordered tile pair (a<=b) of 16x16 spatial tiles.
// 16 waves; wave w computes output row w of both tiles with chained
// v_wmma_f32_16x16x4_f32 (W7 * x  +  W8 * x^T via LDS transpose reads).
// LDS tiles padded to stride 17 -> bank-conflict-free row AND column reads.
// Epilogue: b128 loads of folded term arrays, no exec-divergent branches.
// ---------------------------------------------------------------------------
__global__ void __launch_bounds__(512)
main_kernel(const float* __restrict__ x, const int* __restrict__ mask,
            const float* __restrict__ W,
            const float* __restrict__ Rrow, const float* __restrict__ Ccol,
            const float* __restrict__ Dg, float* __restrict__ out) {
    __shared__ float sTA[16][16][17];   // x[c][r0+i][s0+j]
    __shared__ float sTB[16][16][17];   // x[c][s0+i][r0+j]
    __shared__ int   sMA[16][16];
    __shared__ int   sMB[16][16];

    // blockIdx -> (a, b) with a <= b  (scalar loop, <=128 iters)
    int idx = blockIdx.x;
    int a = 0;
    while (idx >= TTILES - a) { idx -= TTILES - a; ++a; }
    const int b = a + idx;
    const int r0 = a * 16, s0 = b * 16;
    const bool diagpair = (a == b);
    const int tid = threadIdx.x;

    // stage both tiles (all 16 channels) + mask tiles into LDS
    for (int e = tid; e < 16 * 256; e += 512) {
        const int c = e >> 8, rm = e & 255, i = rm >> 4, j = rm & 15;
        sTA[c][i][j] = x[(size_t)c * NELEM + (size_t)(r0 + i) * NSIDE + (s0 + j)];
        sTB[c][i][j] = x[(size_t)c * NELEM + (size_t)(s0 + i) * NSIDE + (r0 + j)];
    }
    if (tid < 256) {
        const int i = tid >> 4, j = tid & 15;
        sMA[i][j] = mask[(r0 + i) * NSIDE + (s0 + j)];
        sMB[i][j] = mask[(s0 + i) * NSIDE + (r0 + j)];
    }
    __syncthreads();

    const int w    = tid >> 5;      // wave id == row index within tile
    const int lane = tid & 31;
    const int h    = lane >> 4;     // lane half (K-split for A/B, M-split for D)
    const int jl   = lane & 15;     // A: M index; B/D: N index
    const float* W7 = W + 7 * 256;
    const float* W8 = W + 8 * 256;

    // A-matrix regs: A[M=o][K=c] = W[c][o]; lanes0-15 K={4t,4t+1}, lanes16-31 K={4t+2,4t+3}
    v2f a7[4], a8[4];
#pragma unroll
    for (int t = 0; t < 4; ++t) {
        const int k0 = 4 * t + 2 * h;
        a7[t].x = W7[k0 * 16 + jl];  a7[t].y = W7[(k0 + 1) * 16 + jl];
        a8[t].x = W8[k0 * 16 + jl];  a8[t].y = W8[(k0 + 1) * 16 + jl];
    }

    // ---- tile 1: outputs at rows r0+w, cols s0+jl ----
    v8f acc = {0.f, 0.f, 0.f, 0.f, 0.f, 0.f, 0.f, 0.f};
#pragma unroll
    for (int t = 0; t < 4; ++t) {                     // W7 * x[:, r, s]
        const int k0 = 4 * t + 2 * h;
        v2f bm; bm.x = sTA[k0][w][jl]; bm.y = sTA[k0 + 1][w][jl];
        acc = __builtin_amdgcn_wmma_f32_16x16x4_f32(false, a7[t], false, bm,
                                                    (short)0, acc, false, false);
    }
#pragma unroll
    for (int t = 0; t < 4; ++t) {                     // W8 * x[:, s, r] (transpose read)
        const int k0 = 4 * t + 2 * h;
        v2f bm; bm.x = sTB[k0][jl][w]; bm.y = sTB[k0 + 1][jl][w];
        acc = __builtin_amdgcn_wmma_f32_16x16x4_f32(false, a8[t], false, bm,
                                                    (short)0, acc, false, false);
    }
    {
        const int r = r0 + w, s = s0 + jl;
        const float m = (float)sMA[w][jl];
        // vectorized epilogue term loads: 8 consecutive o values per lane
        float Rt[8], Ct[8], Dt[8];
        *reinterpret_cast<float4*>(&Rt[0]) = *reinterpret_cast<const float4*>(&Rrow[r * NCH + 8 * h]);
        *reinterpret_cast<float4*>(&Rt[4]) = *reinterpret_cast<const float4*>(&Rrow[r * NCH + 8 * h + 4]);
        *reinterpret_cast<float4*>(&Ct[0]) = *reinterpret_cast<const float4*>(&Ccol[s * NCH + 8 * h]);
        *reinterpret_cast<float4*>(&Ct[4]) = *reinterpret_cast<const float4*>(&Ccol[s * NCH + 8 * h + 4]);
#pragma unroll
        for (int v = 0; v < 8; ++v) Dt[v] = 0.f;
        if (diagpair) {                               // block-uniform branch
            const float dsel = (w == jl) ? 1.f : 0.f; // lane-varying predicate as multiplier
            float Dl[8];
            *reinterpret_cast<float4*>(&Dl[0]) = *reinterpret_cast<const float4*>(&Dg[r * NCH + 8 * h]);
            *reinterpret_cast<float4*>(&Dl[4]) = *reinterpret_cast<const float4*>(&Dg[r * NCH + 8 * h + 4]);
#pragma unroll
            for (int v = 0; v < 8; ++v) Dt[v] = Dl[v] * dsel;
        }
#pragma unroll
        for (int v = 0; v < 8; ++v) {
            const int o = v + 8 * h;                  // D layout: VGPR v -> M=v (+8 upper half)
            const float val = acc[v] + Rt[v] + Ct[v] + Dt[v];
            out[(size_t)o * NELEM + (size_t)r * NSIDE + s] = val * m;
        }
    }

    if (!diagpair) {
        // ---- tile 2: outputs at rows s0+w, cols r0+jl (uniform branch; EXEC all-ones)
        v8f acc2 = {0.f, 0.f, 0.f, 0.f, 0.f, 0.f, 0.f, 0.f};
#pragma unroll
        for (int t = 0; t < 4; ++t) {
            const int k0 = 4 * t + 2 * h;
            v2f bm; bm.x = sTB[k0][w][jl]; bm.y = sTB[k0 + 1][w][jl];
            acc2 = __builtin_amdgcn_wmma_f32_16x16x4_f32(false, a7[t], false, bm,
                                                         (short)0, acc2, false, false);
        }
#pragma unroll
        for (int t = 0; t < 4; ++t) {
            const int k0 = 4 * t + 2 * h;
            v2f bm; bm.x = sTA[k0][jl][w]; bm.y = sTA[k0 + 1][jl][w];
            acc2 = __builtin_amdgcn_wmma_f32_16x16x4_f32(false, a8[t], false, bm,
                                                         (short)0, acc2, false, false);
        }
        const int r = s0 + w, s = r0 + jl;            // r != s always here
        const float m = (float)sMB[w][jl];
        float Rt[8], Ct[8];
        *reinterpret_cast<float4*>(&Rt[0]) = *reinterpret_cast<const float4*>(&Rrow[r * NCH + 8 * h]);
        *reinterpret_cast<float4*>(&Rt[4]) = *reinterpret_cast<const float4*>(&Rrow[r * NCH + 8 * h + 4]);
        *reinterpret_cast<float4*>(&Ct[0]) = *reinterpret_cast<const float4*>(&Ccol[s * NCH + 8 * h]);
        *reinterpret_cast<float4*>(&Ct[4]) = *reinterpret_cast<const float4*>(&Ccol[s * NCH + 8 * h + 4]);
#pragma unroll
        for (int v = 0; v < 8; ++v) {
            const int o = v + 8 * h;
            const float val = acc2[v] + Rt[v] + Ct[v];
            out[(size_t)o * NELEM + (size_t)r * NSIDE + s] = val * m;
        }
    }
}

// ---------------------------------------------------------------------------
extern "C" void kernel_launch(void* const* d_in, const int* in_sizes, int n_in,
                              void* d_out, int out_size, void* d_ws, size_t ws_size,
                              hipStream_t stream) {
    (void)in_sizes; (void)n_in; (void)out_size; (void)ws_size;
    const float* x    = (const float*)d_in[0];
    const int*   msk  = (const int*)d_in[1];
    const float* W    = (const float*)d_in[2];
    const float* bias = (const float*)d_in[3];
    float* out = (float*)d_out;
    float* ws  = (float*)d_ws;

    // workspace layout (floats); total ~1.84 MB
    float* diag_t   = ws;                 // 2048*16
    float* rowsum_t = ws + 32768;         // 2048*16
    float* colsum_t = ws + 65536;         // 2048*16
    float* cpart    = ws + 98304;         // 8*2048*16
    float* Rrow     = ws + 360448;        // 2048*16
    float* Ccol     = ws + 393216;        // 2048*16
    float* Dg       = ws + 425984;        // 2048*16
    float* scal     = ws + 458752;        // 32

    rowsum_diag_kernel<<<16 * 256, 256, 0, stream>>>(x, rowsum_t, diag_t);
    colsum_kernel<<<16 * 8 * 8, 256, 0, stream>>>(x, cpart);
    colsum_fin_kernel<<<128, 256, 0, stream>>>(cpart, colsum_t);
    scal_kernel<<<1, 256, 0, stream>>>(diag_t, rowsum_t, W, bias, scal);
    terms_kernel<<<128, 256, 0, stream>>>(diag_t, rowsum_t, colsum_t, W, scal, Rrow, Ccol, Dg);
    main_kernel<<<NPAIRS, 512, 0, stream>>>(x, msk, W, Rrow, Ccol, Dg, out);
}